// GNNModel_83434034692484
// MI455X (gfx1250) — compile-verified
//
#include <hip/hip_runtime.h>
#include <hip/hip_bf16.h>

// GCN 2-layer: N nodes, E edges, F_IN=128, F_HID=16, F_OUT=1
// deg/norm -> h = x@W1 (WMMA f32 16x16x4) -> edge scatter (L2 atomics)
// -> relu+bias+@W2 per node -> edge scatter scalar -> out

#define F_IN  128
#define F_HID 16

typedef float v2f __attribute__((ext_vector_type(2)));
typedef float v8f __attribute__((ext_vector_type(8)));

// ---------------- kernel 1: init deg=1 (self-loop), agg=0 ----------------
__global__ void k_init(float* __restrict__ deg, float* __restrict__ agg, int n) {
    int i = blockIdx.x * blockDim.x + threadIdx.x;
    if (i < n * F_HID) agg[i] = 0.0f;
    if (i < n)         deg[i] = 1.0f;       // self-loop contributes 1
}

// ---------------- kernel 2: deg[dst] += 1 per edge ----------------
__global__ void k_deg(const long long* __restrict__ dst, float* __restrict__ deg, int e) {
    int i = blockIdx.x * blockDim.x + threadIdx.x;
    if (i >= e) return;
    atomicAdd(&deg[(int)dst[i]], 1.0f);
}

// ---------------- kernel 3: dis = rsqrt(deg)  (deg >= 1 always) ----------------
__global__ void k_dis(const float* __restrict__ deg, float* __restrict__ dis, int n) {
    int i = blockIdx.x * blockDim.x + threadIdx.x;
    if (i >= n) return;
    dis[i] = rsqrtf(deg[i]);
}

// ---------------- kernel 4: h = x @ W1 via V_WMMA_F32_16X16X4_F32 ----------------
// One wave32 handles a 16-row tile. A: 16x4 f32 (2 VGPR), B: 4x16 (2 VGPR),
// C/D: 16x16 f32 (8 VGPR). K = 128 in steps of 4 -> 32 WMMAs per wave.
__global__ void k_gemm1_wmma(const float* __restrict__ x,
                             const float* __restrict__ W1,
                             float* __restrict__ h, int nTiles) {
    int wave = (int)((blockIdx.x * blockDim.x + threadIdx.x) >> 5);
    if (wave >= nTiles) return;                 // wave-uniform: EXEC stays all-1s
    int lane = threadIdx.x & 31;
    int m    = lane & 15;                       // M for A-frag, N for B-frag
    int kh   = (lane >> 4) << 1;                // K-half offset: 0 (lanes 0-15) / 2 (16-31)

    const float* __restrict__ xrow = x + (size_t)(wave * 16 + m) * F_IN;
    v8f c = {};
    #pragma unroll 4
    for (int k0 = 0; k0 < F_IN; k0 += 4) {
        v2f a, b;
        a.x = xrow[k0 + kh];
        a.y = xrow[k0 + kh + 1];
        b.x = W1[(k0 + kh)     * F_HID + m];
        b.y = W1[(k0 + kh + 1) * F_HID + m];
        c = __builtin_amdgcn_wmma_f32_16x16x4_f32(
                /*neg_a=*/false, a, /*neg_b=*/false, b,
                /*c_mod=*/(short)0, c, /*reuse_a=*/false, /*reuse_b=*/false);
    }
    // C/D layout: VGPR v, lane l -> element [M = v + 8*(l>>4)][N = l&15]
    float* __restrict__ hrow = h + (size_t)wave * 16 * F_HID;
    int half = lane >> 4;
    #pragma unroll
    for (int v = 0; v < 8; ++v)
        hrow[(v + half * 8) * F_HID + m] = c[v];
}

// tail rows (n % 16) — scalar fallback, launched only if needed
__global__ void k_gemm1_tail(const float* __restrict__ x, const float* __restrict__ W1,
                             float* __restrict__ h, int row0, int nrows) {
    int t = blockIdx.x * blockDim.x + threadIdx.x;
    if (t >= nrows * F_HID) return;
    int r = row0 + t / F_HID, j = t % F_HID;
    float acc = 0.f;
    for (int k = 0; k < F_IN; ++k) acc += x[(size_t)r * F_IN + k] * W1[k * F_HID + j];
    h[(size_t)r * F_HID + j] = acc;
}

// ---------------- kernel 5: layer-1 edge scatter. 16 lanes per edge ----------------
__global__ void k_scatter1(const long long* __restrict__ src,
                           const long long* __restrict__ dst,
                           const float* __restrict__ dis,
                           const float* __restrict__ h,
                           float* __restrict__ agg, int e) {
    long long t = (long long)blockIdx.x * blockDim.x + threadIdx.x;
    int ei = (int)(t >> 4);
    if (ei >= e) return;
    int j = (int)(t & 15);
    int s = (int)src[ei], d = (int)dst[ei];
    float norm = dis[s] * dis[d];
    atomicAdd(&agg[(size_t)d * F_HID + j], h[(size_t)s * F_HID + j] * norm);
}

// ---------------- kernel 6: per-node relu(agg + self + b1) @ W2; seed out ----------------
__global__ void k_node(const float* __restrict__ agg, const float* __restrict__ h,
                       const float* __restrict__ dis,
                       const float* __restrict__ b1, const float* __restrict__ W2,
                       const float* __restrict__ b2,
                       float* __restrict__ z, float* __restrict__ out, int n) {
    int i = blockIdx.x * blockDim.x + threadIdx.x;
    if (i >= n) return;
    float dn = dis[i];
    float self = dn * dn;                       // self-loop norm = dis[i]^2
    const float4* ag = (const float4*)(agg + (size_t)i * F_HID);
    const float4* hr = (const float4*)(h   + (size_t)i * F_HID);
    const float4* bb = (const float4*)b1;
    const float4* ww = (const float4*)W2;
    float acc = 0.f;
    #pragma unroll
    for (int q = 0; q < 4; ++q) {
        float4 a = ag[q], hv = hr[q], b = bb[q], w = ww[q];
        acc += fmaxf(fmaf(hv.x, self, a.x) + b.x, 0.f) * w.x;
        acc += fmaxf(fmaf(hv.y, self, a.y) + b.y, 0.f) * w.y;
        acc += fmaxf(fmaf(hv.z, self, a.z) + b.z, 0.f) * w.z;
        acc += fmaxf(fmaf(hv.w, self, a.w) + b.w, 0.f) * w.w;
    }
    z[i]   = acc;
    out[i] = b2[0] + acc * self;                // layer-2 bias + self-loop term
}

// ---------------- kernel 7: layer-2 edge scatter, scalar ----------------
__global__ void k_scatter2(const long long* __restrict__ src,
                           const long long* __restrict__ dst,
                           const float* __restrict__ dis,
                           const float* __restrict__ z,
                           float* __restrict__ out, int e) {
    int i = blockIdx.x * blockDim.x + threadIdx.x;
    if (i >= e) return;
    int s = (int)src[i], d = (int)dst[i];
    atomicAdd(&out[d], z[s] * dis[s] * dis[d]);
}

static inline size_t alignup(size_t v) { return (v + 255) & ~(size_t)255; }

extern "C" void kernel_launch(void* const* d_in, const int* in_sizes, int n_in,
                              void* d_out, int out_size, void* d_ws, size_t ws_size,
                              hipStream_t stream) {
    const float*     x   = (const float*)d_in[0];
    const long long* ei  = (const long long*)d_in[1];
    const float*     W1  = (const float*)d_in[2];
    const float*     b1  = (const float*)d_in[3];
    const float*     W2  = (const float*)d_in[4];
    const float*     b2  = (const float*)d_in[5];
    float*           out = (float*)d_out;

    const int n = in_sizes[0] / F_IN;           // 100000
    const int e = in_sizes[1] / 2;              // 3200000
    const long long* src = ei;
    const long long* dst = ei + e;

    // workspace carve: deg, dis, h[N,16], agg[N,16], z  (~13.6 MB)
    char* ws = (char*)d_ws;
    float* deg = (float*)ws;  ws += alignup((size_t)n * 4);
    float* dis = (float*)ws;  ws += alignup((size_t)n * 4);
    float* h   = (float*)ws;  ws += alignup((size_t)n * F_HID * 4);
    float* agg = (float*)ws;  ws += alignup((size_t)n * F_HID * 4);
    float* z   = (float*)ws;  ws += alignup((size_t)n * 4);
    (void)ws_size; (void)n_in; (void)out_size;

    const int B = 256;
    // 1. init
    k_init<<<(n * F_HID + B - 1) / B, B, 0, stream>>>(deg, agg, n);
    // 2. degree accumulation
    k_deg<<<(e + B - 1) / B, B, 0, stream>>>(dst, deg, e);
    // 3. dis = rsqrt(deg)
    k_dis<<<(n + B - 1) / B, B, 0, stream>>>(deg, dis, n);
    // 4. h = x @ W1 (WMMA), one wave32 per 16-row tile
    int nTiles = n / 16;
    int gemmThreads = nTiles * 32;
    k_gemm1_wmma<<<(gemmThreads + B - 1) / B, B, 0, stream>>>(x, W1, h, nTiles);
    int rem = n % 16;
    if (rem > 0)
        k_gemm1_tail<<<(rem * F_HID + B - 1) / B, B, 0, stream>>>(x, W1, h, n - rem, rem);
    // 5. layer-1 scatter: 16 lanes per edge
    long long s1t = (long long)e * F_HID;
    k_scatter1<<<(unsigned)((s1t + B - 1) / B), B, 0, stream>>>(src, dst, dis, h, agg, e);
    // 6. per-node: relu(agg + self + b1) @ W2, seed out with b2 + self-loop
    k_node<<<(n + B - 1) / B, B, 0, stream>>>(agg, h, dis, b1, W2, b2, z, out, n);
    // 7. layer-2 scatter
    k_scatter2<<<(e + B - 1) / B, B, 0, stream>>>(src, dst, dis, z, out, e);
}